// QuantizerTl_54709293416649
// MI455X (gfx1250) — compile-verified
//
#include <hip/hip_runtime.h>
#include <stdint.h>

#define H           4096
#define TPB         256
#define PER_LANE    16          // floats handled per thread per row
#define ROWS_PER_WG 4           // 16384 rows / 4 = 4096 workgroups
#define NWAVES      (TPB / 32)

__global__ __launch_bounds__(TPB) void quant_int4_rowwise(
    const float* __restrict__ x,
    uint8_t*     __restrict__ qout,
    float*       __restrict__ sout)
{
    __shared__ float lds[ROWS_PER_WG][H];   // 4 x 16 KB staging buffers (64 KB)
    __shared__ float wmax[NWAVES];

    const int tid  = threadIdx.x;
    const int wid  = tid >> 5;
    const int lane = tid & 31;
    const int row0 = blockIdx.x * ROWS_PER_WG;

    // ---- Issue async DMA (global -> LDS) for ALL rows up front ------------
    // Each lane: 4 x b128 per row => 16 async instructions per wave.
    // Rows 1..3 stream in from HBM while earlier rows are processed.
    #pragma unroll
    for (int r = 0; r < ROWS_PER_WG; ++r) {
        const uint64_t sbase = (uint64_t)(uintptr_t)(x + (size_t)(row0 + r) * H);
        #pragma unroll
        for (int j = 0; j < 4; ++j) {
            const int      chunk = tid + j * TPB;          // 16-byte chunk id
            const uint32_t goff  = (uint32_t)chunk * 16u;  // global byte offset
            // Low 32 bits of a generic LDS pointer == LDS byte offset (flat aperture rule)
            const uint32_t ldsa  = (uint32_t)(uintptr_t)&lds[r][chunk * 4];
            asm volatile("global_load_async_to_lds_b128 %0, %1, %2"
                         :
                         : "v"(ldsa), "v"(goff), "s"(sbase)
                         : "memory");
        }
    }

    #pragma unroll
    for (int r = 0; r < ROWS_PER_WG; ++r) {
        // Async loads retire in issue order: cnt<=(3-r)*4 means row r's 4 are done.
        if      (r == 0) asm volatile("s_wait_asynccnt 12" ::: "memory");
        else if (r == 1) asm volatile("s_wait_asynccnt 8"  ::: "memory");
        else if (r == 2) asm volatile("s_wait_asynccnt 4"  ::: "memory");
        else             asm volatile("s_wait_asynccnt 0"  ::: "memory");
        __syncthreads();   // all waves' DMA for this row is now visible in LDS

        // ---- Read my 16 contiguous floats back from LDS (ds_load_b128 x4) --
        const float4* lp = reinterpret_cast<const float4*>(&lds[r][tid * PER_LANE]);
        float4 f0 = lp[0], f1 = lp[1], f2 = lp[2], f3 = lp[3];
        float v[PER_LANE] = { f0.x, f0.y, f0.z, f0.w,
                              f1.x, f1.y, f1.z, f1.w,
                              f2.x, f2.y, f2.z, f2.w,
                              f3.x, f3.y, f3.z, f3.w };

        // ---- Per-lane absmax, then wave32 + cross-wave reduction -----------
        float m = 0.0f;
        #pragma unroll
        for (int k = 0; k < PER_LANE; ++k) m = fmaxf(m, fabsf(v[k]));

        #pragma unroll
        for (int off = 16; off >= 1; off >>= 1)
            m = fmaxf(m, __shfl_xor(m, off, 32));

        if (lane == 0) wmax[wid] = m;
        __syncthreads();

        float am = wmax[0];
        #pragma unroll
        for (int w = 1; w < NWAVES; ++w) am = fmaxf(am, wmax[w]);

        // scale = max(absmax * CLIP_RATIO / QMAX, 1e-8), CLIP_RATIO=1, QMAX=7
        const float scale = fmaxf(am * (1.0f / 7.0f), 1e-8f);
        const float inv   = 1.0f / scale;

        // ---- Quantize to int4 (RN-even like jnp.round), pack 2/byte --------
        uint64_t p = 0;
        #pragma unroll
        for (int k = 0; k < 8; ++k) {
            float t0 = fminf(fmaxf(__builtin_rintf(v[2 * k]     * inv), -8.0f), 7.0f);
            float t1 = fminf(fmaxf(__builtin_rintf(v[2 * k + 1] * inv), -8.0f), 7.0f);
            uint32_t b = ((uint32_t)(int)t0 & 0xFu) | (((uint32_t)(int)t1 & 0xFu) << 4);
            p |= (uint64_t)b << (8 * k);
        }

        const size_t row = (size_t)(row0 + r);
        uint64_t* dst = reinterpret_cast<uint64_t*>(qout + row * (H / 2)) + tid;
        __builtin_nontemporal_store(p, dst);   // streaming store, 8 B/lane
        if (tid == 0) sout[row] = scale;
    }
}

extern "C" void kernel_launch(void* const* d_in, const int* in_sizes, int n_in,
                              void* d_out, int out_size, void* d_ws, size_t ws_size,
                              hipStream_t stream)
{
    const float* x = (const float*)d_in[0];
    const int total = in_sizes[0];        // 4 * 4096 * 4096
    const int nrows = total / H;          // 16384

    // Outputs concatenated flat in return order:
    //   [0, nrows*H/2) bytes : packed int4 pairs (int8)
    //   then nrows floats    : per-row scales
    uint8_t* qout = (uint8_t*)d_out;
    float*   sout = (float*)(qout + (size_t)nrows * (H / 2));

    const int blocks = nrows / ROWS_PER_WG;
    quant_int4_rowwise<<<blocks, TPB, 0, stream>>>(x, qout, sout);
}